// autoencoder_54735063220231
// MI455X (gfx1250) — compile-verified
//
#include <hip/hip_runtime.h>
#include <math.h>

typedef float v2f __attribute__((ext_vector_type(2)));
typedef float v8f __attribute__((ext_vector_type(8)));

#define NB 256
#define NPAIR 32640   // 256*255/2

// ---------------------------------------------------------------------------
// GEMV: out[r] = dot(W[r, :], x) + bias[r]  (optional sigmoid, optional 2nd sink)
// One wave32 per row; lanes stream float4 (global_load_b128), shuffle-reduce.
// Memory bound: this is the 201MB weight stream, ~8.7us floor at 23.3 TB/s.
// ---------------------------------------------------------------------------
template <int SIG>
__global__ void gemv_kernel(const float* __restrict__ W,
                            const float* __restrict__ x,
                            const float* __restrict__ bias,
                            float* __restrict__ out,
                            float* __restrict__ out2,
                            int cols) {
    const int wave = threadIdx.x >> 5;      // 8 waves / block
    const int lane = threadIdx.x & 31;
    const int row  = blockIdx.x * 8 + wave;

    const float4* __restrict__ Wr = (const float4*)(W + (size_t)row * cols);
    const float4* __restrict__ xv = (const float4*)x;
    const int n4 = cols >> 2;

    float acc = 0.0f;
    for (int i = lane; i < n4; i += 32) {
        float4 w = Wr[i];
        float4 v = xv[i];
        acc += w.x * v.x + w.y * v.y + w.z * v.z + w.w * v.w;
    }
    // wave32 butterfly reduction
    #pragma unroll
    for (int off = 16; off > 0; off >>= 1)
        acc += __shfl_xor(acc, off, 32);

    if (lane == 0) {
        float v = acc + bias[row];
        if (SIG) v = 1.0f / (1.0f + __expf(-v));
        out[row] = v;
        if (out2) out2[row] = v;
    }
}

// ---------------------------------------------------------------------------
// Gram matrix G = R * R^T for R: [256 x K] via V_WMMA_F32_16X16X4_F32.
// One wave per 16x16 output tile; 16x16 tile grid -> 256 single-wave blocks.
// A (16x4 f32) layout: lane l holds M = l%16, K-pair base = (l/16)*2 in 2 VGPRs
// (consecutive K -> contiguous float2 load). B (4x16) is the mirrored layout,
// and B[k][n] = R[n][k] so it is also a contiguous float2 of row n.
// C/D (16x16 f32): VGPR r, lane l -> M = r + (l/16)*8, N = l%16.
// ---------------------------------------------------------------------------
template <int K>
__global__ void gram_kernel(const float* __restrict__ R, float* __restrict__ G) {
    const int tile = blockIdx.x;            // 0..255
    const int tm = tile >> 4;
    const int tn = tile & 15;
    const int lane = threadIdx.x;           // blockDim.x == 32
    const int l16  = lane & 15;
    const int koff = (lane >> 4) * 2;

    const float* __restrict__ Arow = R + (size_t)(tm * 16 + l16) * K;
    const float* __restrict__ Brow = R + (size_t)(tn * 16 + l16) * K;

    v8f acc = {};
    #pragma unroll
    for (int kb = 0; kb < K; kb += 4) {
        v2f a, b;
        a.x = Arow[kb + koff + 0];
        a.y = Arow[kb + koff + 1];
        b.x = Brow[kb + koff + 0];
        b.y = Brow[kb + koff + 1];
        acc = __builtin_amdgcn_wmma_f32_16x16x4_f32(
            /*neg_a=*/false, a, /*neg_b=*/false, b,
            /*c_mod=*/(short)0, acc, /*reuse_a=*/false, /*reuse_b=*/false);
    }

    const int mbase = tm * 16 + (lane >> 4) * 8;
    const int col   = tn * 16 + l16;
    #pragma unroll
    for (int r = 0; r < 8; ++r)
        G[(size_t)(mbase + r) * NB + col] = acc[r];
}

// ---------------------------------------------------------------------------
// p -> (i, j), i < j, row-major triu order. start(i) = i*(511-i)/2 for B=256.
// ---------------------------------------------------------------------------
__device__ __forceinline__ void pair_ij(int p, int& io, int& jo) {
    float disc = 511.0f * 511.0f - 8.0f * (float)p;
    int i = (int)((511.0f - sqrtf(disc)) * 0.5f);
    if (i < 0) i = 0;
    if (i > 254) i = 254;
    while (i < 254 && ((i + 1) * (511 - (i + 1)) / 2) <= p) ++i;
    while (i > 0 && (i * (511 - i) / 2) > p) --i;
    io = i;
    jo = p - i * (511 - i) / 2 + i + 1;
}

// dist²(i,j) = G[i,i] + G[j,j] - 2 G[i,j]  (clamped at 0)
__global__ void pairdist_kernel(const float* __restrict__ G, float* __restrict__ outd) {
    int p = blockIdx.x * blockDim.x + threadIdx.x;
    if (p >= NPAIR) return;
    int i, j;
    pair_ij(p, i, j);
    float s = G[(size_t)i * NB + i] + G[(size_t)j * NB + j]
            - 2.0f * G[(size_t)i * NB + j];
    outd[p] = fmaxf(s, 0.0f);
}

// out_diff over first 256 scalars of the flattened output
__global__ void outdiff_kernel(const float* __restrict__ o, float* __restrict__ outd) {
    int p = blockIdx.x * blockDim.x + threadIdx.x;
    if (p >= NPAIR) return;
    int i, j;
    pair_ij(p, i, j);
    float d = o[i] - o[j];
    outd[p] = d * d;
}

// ---------------------------------------------------------------------------
extern "C" void kernel_launch(void* const* d_in, const int* in_sizes, int n_in,
                              void* d_out, int out_size, void* d_ws, size_t ws_size,
                              hipStream_t stream) {
    const float* x     = (const float*)d_in[0];   // [4096]
    const float* enc_w = (const float*)d_in[1];   // [4096,4096]
    const float* enc_b = (const float*)d_in[2];   // [4096]
    const float* fc_w  = (const float*)d_in[3];   // [2048,4096]
    const float* fc_b  = (const float*)d_in[4];   // [2048]
    const float* tr_w  = (const float*)d_in[5];   // [4096,2048]
    const float* tr_b  = (const float*)d_in[6];   // [4096]
    const float* dec_w = (const float*)d_in[7];   // [4096,4096]
    const float* dec_b = (const float*)d_in[8];   // [4096]

    float* ws = (float*)d_ws;
    float* z1    = ws;             // [4096]
    float* z     = ws + 4096;      // [2048]
    float* y     = ws + 6144;      // [4096]
    float* G_in  = ws + 10240;     // [256*256]
    float* G_lat = ws + 75776;     // [256*256]

    float* out      = (float*)d_out;        // [4096]
    float* in_diff  = out + 4096;           // [32640]
    float* lat_diff = in_diff + NPAIR;      // [32640]
    float* out_diff = lat_diff + NPAIR;     // [32640]
    float* lat_repr = out_diff + NPAIR;     // [2048]

    // Chained GEMVs (stream order enforces dependencies)
    gemv_kernel<0><<<512, 256, 0, stream>>>(enc_w, x,  enc_b, z1, nullptr,  4096);
    gemv_kernel<0><<<256, 256, 0, stream>>>(fc_w,  z1, fc_b,  z,  lat_repr, 4096);
    gemv_kernel<0><<<512, 256, 0, stream>>>(tr_w,  z,  tr_b,  y,  nullptr,  2048);
    gemv_kernel<1><<<512, 256, 0, stream>>>(dec_w, y,  dec_b, out, nullptr, 4096);

    // Gram matrices via f32 WMMA (one wave per 16x16 tile)
    gram_kernel<16><<<256, 32, 0, stream>>>(x, G_in);       // in_data [256x16]
    gram_kernel<8><<<256, 32, 0, stream>>>(z, G_lat);       // lat_repr [256x8]

    // Pairwise reductions
    pairdist_kernel<<<(NPAIR + 255) / 256, 256, 0, stream>>>(G_in,  in_diff);
    pairdist_kernel<<<(NPAIR + 255) / 256, 256, 0, stream>>>(G_lat, lat_diff);
    outdiff_kernel <<<(NPAIR + 255) / 256, 256, 0, stream>>>(out,   out_diff);
}